// IntSoftMax_18167711662302
// MI455X (gfx1250) — compile-verified
//
#include <hip/hip_runtime.h>
#include <hip/hip_bf16.h>

// IntSoftMax on MI455X (gfx1250): bandwidth-bound (1.61 GB @ 23.3 TB/s ~= 69us floor).
// One row (2048 fp32, 8 KB) per 256-thread block (8 wave32 waves).
// Data path: global -(async NT)-> LDS -> registers -(NT store)-> global.
// Exactly one HBM read and one HBM write per element; NT hints keep the
// 1.6 GB stream from thrashing the 192 MB L2 (zero reuse workload).

#define N_SHIFT 15
#define SLEN    2048
#define TPB     256
#define EPT     8            // elements per thread
#define NWAVES  (TPB / 32)

typedef __attribute__((ext_vector_type(4))) float v4f;   // native vector: OK for NT builtins

__device__ __forceinline__ void async_load_b128_to_lds(unsigned lds_byte_off,
                                                       unsigned long long gaddr) {
  // GV mode: per-lane 64-bit global address, per-lane LDS byte offset.
  // Non-temporal: row is consumed once, don't cache it.
  asm volatile("global_load_async_to_lds_b128 %0, %1, off th:TH_LOAD_NT"
               :
               : "v"(lds_byte_off), "v"(gaddr)
               : "memory");
}

__device__ __forceinline__ void wait_async0() {
#if __has_builtin(__builtin_amdgcn_s_wait_asynccnt)
  __builtin_amdgcn_s_wait_asynccnt(0);
#else
  asm volatile("s_wait_asynccnt 0" ::: "memory");
#endif
}

__global__ __launch_bounds__(TPB) void intsoftmax_kernel(
    const float* __restrict__ x, const float* __restrict__ s_pre,
    float* __restrict__ out, int rows) {
  __shared__ float buf[SLEN];        // 8 KB row staging (async-copy target)
  __shared__ float redmax[NWAVES];
  __shared__ float redsum[NWAVES];

  const int row = blockIdx.x;
  if (row >= rows) return;
  const long long base = (long long)row * SLEN;
  const int t    = threadIdx.x;
  const int wave = t >> 5;
  const int lane = t & 31;

  const float s      = s_pre[0];
  const float inv_s  = 1.0f / s;
  const float x0     = floorf(-1.0f / s);     // negative
  const float inv_x0 = 1.0f / x0;
  const float nx0    = (float)N_SHIFT * x0;

  // ---- async-stage the row into LDS: two contiguous 4 KB chunks ----
  {
    unsigned long long g0 = (unsigned long long)(x + base + t * 4);
    unsigned long long g1 = (unsigned long long)(x + base + 1024 + t * 4);
    unsigned l0 = (unsigned)(uintptr_t)(&buf[t * 4]);          // low 32 bits = LDS offset
    unsigned l1 = (unsigned)(uintptr_t)(&buf[1024 + t * 4]);
    async_load_b128_to_lds(l0, g0);
    async_load_b128_to_lds(l1, g1);
    wait_async0();   // each lane reads back only what it wrote -> no barrier needed
  }

  v4f a = *(const v4f*)&buf[t * 4];
  v4f b = *(const v4f*)&buf[1024 + t * 4];
  float v[EPT] = {a.x, a.y, a.z, a.w, b.x, b.y, b.z, b.w};

  // ---- row max (wave32 shuffle + cross-wave LDS combine) ----
  float m = -3.402823466e38f;
#pragma unroll
  for (int i = 0; i < EPT; ++i) { v[i] *= inv_s; m = fmaxf(m, v[i]); }
  for (int o = 16; o > 0; o >>= 1) m = fmaxf(m, __shfl_xor(m, o, 32));
  if (lane == 0) redmax[wave] = m;
  __syncthreads();
  m = redmax[0];
#pragma unroll
  for (int j = 1; j < NWAVES; ++j) m = fmaxf(m, redmax[j]);

  // ---- shift-based exp approximation + row sum ----
  float sum = 0.0f;
#pragma unroll
  for (int i = 0; i < EPT; ++i) {
    float xi = v[i] - m;                                   // <= 0
    xi = xi + floorf(xi * 0.5f) - floorf(xi * 0.0625f);    // x + x>>1 - x>>4
    xi = fmaxf(xi, nx0);
    float q = floorf(xi * inv_x0);                         // q in [0, 15]
    float r = xi - x0 * q;
    float e = fmaf(r, 0.5f, -x0);
    e = fmaxf(floorf(ldexpf(e, N_SHIFT - (int)q)), 0.0f);
    v[i] = e;
    sum += e;
  }
  for (int o = 16; o > 0; o >>= 1) sum += __shfl_xor(sum, o, 32);
  if (lane == 0) redsum[wave] = sum;
  __syncthreads();
  sum = 0.0f;
#pragma unroll
  for (int j = 0; j < NWAVES; ++j) sum += redsum[j];

  // ---- fixed-point renormalization ----
  sum = fminf(sum, 2147483647.0f);
  const float factor = floorf(2147483647.0f / sum);
  const float c      = factor * (1.0f / 16777216.0f);      // factor / 2^(31-8+1)
  const float out_sf = 0.0078125f;                         // 1 / 2^7

#pragma unroll
  for (int i = 0; i < EPT; ++i) v[i] = floorf(v[i] * c) * out_sf;

  v4f oa = {v[0], v[1], v[2], v[3]};
  v4f ob = {v[4], v[5], v[6], v[7]};
  __builtin_nontemporal_store(oa, (v4f*)(out + base + t * 4));
  __builtin_nontemporal_store(ob, (v4f*)(out + base + 1024 + t * 4));

  // trailing scalar output: out_sf
  if (row == 0 && t == 0) out[(long long)rows * SLEN] = out_sf;
}

extern "C" void kernel_launch(void* const* d_in, const int* in_sizes, int n_in,
                              void* d_out, int out_size, void* d_ws, size_t ws_size,
                              hipStream_t stream) {
  (void)n_in; (void)d_ws; (void)ws_size; (void)out_size;
  const float* x     = (const float*)d_in[0];
  const float* s_pre = (const float*)d_in[1];
  float*       out   = (float*)d_out;
  const long long n  = (long long)in_sizes[0];
  const int rows     = (int)(n / SLEN);     // B*H*S = 98304
  intsoftmax_kernel<<<rows, TPB, 0, stream>>>(x, s_pre, out, rows);
}